// GAT_65231963291731
// MI455X (gfx1250) — compile-verified
//
#include <hip/hip_runtime.h>
#include <hip/hip_bf16.h>

#define N_NODES   100000
#define HEADS     4
#define HEAD_DIM  16
#define HC        64
#define OUT_DIM   2
#define N_GRAPHS  512
#define NEDGES    3200000LL
#define ETOT      (NEDGES + (long long)N_NODES)
#define NEG_SLOPE 0.2f

typedef __attribute__((ext_vector_type(2))) float v2f;
typedef __attribute__((ext_vector_type(8))) float v8f;

// Monotone float<->uint key so atomicMax(u32) implements float max for any sign.
__device__ __forceinline__ unsigned fkey(float f) {
    unsigned b = __float_as_uint(f);
    return b ^ ((b & 0x80000000u) ? 0xFFFFFFFFu : 0x80000000u);
}
__device__ __forceinline__ float funkey(unsigned k) {
    unsigned b = (k & 0x80000000u) ? (k ^ 0x80000000u) : ~k;
    return __uint_as_float(b);
}

// ---------------- 0: zero scratch ----------------
__global__ void k_zero(unsigned* __restrict__ p, long long n) {
    long long i = (long long)blockIdx.x * blockDim.x + threadIdx.x;
    if (i < n) p[i] = 0u;
}

// ---------------- 1: h = x @ W via V_WMMA_F32_16X16X4_F32 ----------------
// A (16x4 f32): lanes 0-15 -> M=lane, VGPR0=K0, VGPR1=K1; lanes 16-31 -> M=lane-16, VGPR0=K2, VGPR1=K3.
// K=3 real, K3 zero-padded. B (4x16) mirrored: V0={K0|K2}, V1={K1|K3}.
// D (16x16 f32): VGPR r -> row (lane<16 ? r : r+8), col = lane&15.
__global__ void k_feat_wmma(const float* __restrict__ x, const float* __restrict__ W,
                            float* __restrict__ h, int nTiles) {
    int wave = (int)((blockIdx.x * blockDim.x + threadIdx.x) >> 5);
    int lane = threadIdx.x & 31;
    if (wave >= nTiles) return;            // wave-uniform: EXEC stays all-1s for WMMA
    int row0 = wave * 16;
    int m    = lane & 15;
    int node = row0 + m;
    bool lo  = (lane < 16);

    v2f A;
    if (lo) { A.x = x[node * 3 + 0]; A.y = x[node * 3 + 1]; }
    else    { A.x = x[node * 3 + 2]; A.y = 0.0f; }

    int rbase = lo ? 0 : 8;
#pragma unroll
    for (int head = 0; head < HEADS; ++head) {
        int c = head * 16 + m;             // global output column
        v2f B;
        if (lo) { B.x = W[0 * HC + c]; B.y = W[1 * HC + c]; }
        else    { B.x = W[2 * HC + c]; B.y = 0.0f; }
        v8f C = {};
        C = __builtin_amdgcn_wmma_f32_16x16x4_f32(
                /*neg_a=*/false, A, /*neg_b=*/false, B,
                /*c_mod=*/(short)0, C, /*reuse_a=*/false, /*reuse_b=*/false);
#pragma unroll
        for (int r = 0; r < 8; ++r)
            h[(long long)(row0 + rbase + r) * HC + c] = C[r];
    }
}

// ---------------- 2: per-(node,head) attention logits ----------------
__global__ void k_att(const float* __restrict__ h,
                      const float* __restrict__ att_src, const float* __restrict__ att_dst,
                      float* __restrict__ as, float* __restrict__ ad, int n) {
    int i = blockIdx.x * blockDim.x + threadIdx.x;
    if (i >= n) return;                    // n = N_NODES*HEADS
    int node = i >> 2, head = i & 3;
    const float* hp = h + (long long)node * HC + head * HEAD_DIM;
    float s1 = 0.f, s2 = 0.f;
#pragma unroll
    for (int c = 0; c < HEAD_DIM; ++c) {
        float v = hp[c];
        s1 += v * att_src[head * HEAD_DIM + c];
        s2 += v * att_dst[head * HEAD_DIM + c];
    }
    as[i] = s1; ad[i] = s2;
}

// ---------------- 3: edge pass 1 — segment max over dst ----------------
__global__ void k_edge_max(const long long* __restrict__ ei,
                           const float* __restrict__ as, const float* __restrict__ ad,
                           unsigned* __restrict__ mkey, long long eTot) {
    long long i = (long long)blockIdx.x * blockDim.x + threadIdx.x;
    if (i >= eTot) return;
    int s, d;
    if (i < NEDGES) { s = (int)ei[i]; d = (int)ei[NEDGES + i]; }
    else            { s = d = (int)(i - NEDGES); }           // self loop
#pragma unroll
    for (int hd = 0; hd < HEADS; ++hd) {
        float e = as[s * HEADS + hd] + ad[d * HEADS + hd];
        e = (e > 0.f) ? e : NEG_SLOPE * e;                   // leaky_relu
        atomicMax(&mkey[d * HEADS + hd], fkey(e));
    }
}

// ---------------- 4: edge pass 2 — exp + fused weighted accumulation ----------------
// alpha = p / s with p = exp(e-m); accumulate acc += p*h[src], den += p; normalize later.
__global__ void k_edge_acc(const long long* __restrict__ ei,
                           const float* __restrict__ as, const float* __restrict__ ad,
                           const unsigned* __restrict__ mkey, const float* __restrict__ h,
                           float* __restrict__ den, float* __restrict__ acc, long long eTot) {
    long long i = (long long)blockIdx.x * blockDim.x + threadIdx.x;
    if (i >= eTot) return;
    int s, d;
    if (i < NEDGES) { s = (int)ei[i]; d = (int)ei[NEDGES + i]; }
    else            { s = d = (int)(i - NEDGES); }
#pragma unroll
    for (int hd = 0; hd < HEADS; ++hd) {
        float e = as[s * HEADS + hd] + ad[d * HEADS + hd];
        e = (e > 0.f) ? e : NEG_SLOPE * e;
        float m = funkey(mkey[d * HEADS + hd]);
        float p = __expf(e - m);
        atomicAdd(&den[d * HEADS + hd], p);
        const float* hp = h   + (long long)s * HC + hd * HEAD_DIM;
        float*       ap = acc + (long long)d * HC + hd * HEAD_DIM;
#pragma unroll
        for (int c = 0; c < HEAD_DIM; ++c)
            atomicAdd(&ap[c], p * hp[c]);
    }
}

// ---------------- 5: normalize + bias + relu + graph max-pool ----------------
__global__ void k_out_pool(const float* __restrict__ acc, const float* __restrict__ den,
                           const float* __restrict__ bias, const long long* __restrict__ batch,
                           unsigned* __restrict__ pkey, int total) {
    int i = blockIdx.x * blockDim.x + threadIdx.x;
    if (i >= total) return;                // total = N_NODES*HC
    int node = i >> 6, c = i & 63, head = c >> 4;
    float v = acc[i] / den[node * HEADS + head] + bias[c];
    v = (v > 0.f) ? v : 0.f;
    int g = (int)batch[node];
    atomicMax(&pkey[g * HC + c], fkey(v));
}

// ---------------- 6: classifier pooled @ clf_W + clf_b ----------------
__global__ void k_clf(const unsigned* __restrict__ pkey, const float* __restrict__ clf_W,
                      const float* __restrict__ clf_b, float* __restrict__ out, int total) {
    int i = blockIdx.x * blockDim.x + threadIdx.x;
    if (i >= total) return;                // total = N_GRAPHS*OUT_DIM
    int g = i >> 1, o = i & 1;
    float s = clf_b[o];
#pragma unroll
    for (int c = 0; c < HC; ++c)
        s += funkey(pkey[g * HC + c]) * clf_W[c * OUT_DIM + o];
    out[i] = s;
}

extern "C" void kernel_launch(void* const* d_in, const int* in_sizes, int n_in,
                              void* d_out, int out_size, void* d_ws, size_t ws_size,
                              hipStream_t stream) {
    const float*     x       = (const float*)d_in[0];
    const long long* ei      = (const long long*)d_in[1];
    const long long* batch   = (const long long*)d_in[2];
    const float*     W       = (const float*)d_in[3];
    const float*     att_src = (const float*)d_in[4];
    const float*     att_dst = (const float*)d_in[5];
    const float*     bias    = (const float*)d_in[6];
    const float*     clf_W   = (const float*)d_in[7];
    const float*     clf_b   = (const float*)d_in[8];
    float*           out     = (float*)d_out;

    const long long HN = (long long)N_NODES * HC;      // 6.4M
    const long long AN = (long long)N_NODES * HEADS;   // 0.4M
    const long long PK = (long long)N_GRAPHS * HC;     // 32768

    char* ws = (char*)d_ws;
    float*    h    = (float*)(ws);
    float*    acc  = (float*)(ws + HN * 4);
    float*    as   = (float*)(ws + (HN * 2) * 4);
    float*    ad   = (float*)(ws + (HN * 2 + AN) * 4);
    unsigned* mkey = (unsigned*)(ws + (HN * 2 + AN * 2) * 4);
    float*    den  = (float*)(ws + (HN * 2 + AN * 3) * 4);
    unsigned* pkey = (unsigned*)(ws + (HN * 2 + AN * 4) * 4);

    // Zero acc..pkey (contiguous region): acc, as, ad, mkey, den, pkey.
    long long nz = HN + AN * 4 + PK;
    k_zero<<<(unsigned)((nz + 255) / 256), 256, 0, stream>>>((unsigned*)acc, nz);

    // 1: feature transform (WMMA). 100000/16 = 6250 tiles, 8 waves/block.
    int nTiles = N_NODES / 16;
    k_feat_wmma<<<(nTiles + 7) / 8, 256, 0, stream>>>(x, W, h, nTiles);

    // 2: attention logits per (node, head)
    int nAtt = N_NODES * HEADS;
    k_att<<<(nAtt + 255) / 256, 256, 0, stream>>>(h, att_src, att_dst, as, ad, nAtt);

    // 3: edge segment-max
    unsigned eBlocks = (unsigned)((ETOT + 255) / 256);
    k_edge_max<<<eBlocks, 256, 0, stream>>>(ei, as, ad, mkey, ETOT);

    // 4: edge exp + fused weighted accumulation
    k_edge_acc<<<eBlocks, 256, 0, stream>>>(ei, as, ad, mkey, h, den, acc, ETOT);

    // 5: normalize + relu + graph max-pool
    int nOut = N_NODES * HC;
    k_out_pool<<<(nOut + 255) / 256, 256, 0, stream>>>(acc, den, bias, batch, pkey, nOut);

    // 6: classifier
    int nClf = N_GRAPHS * OUT_DIM;
    k_clf<<<(nClf + 255) / 256, 256, 0, stream>>>(pkey, clf_W, clf_b, out, nClf);
}